// TransitionDown_54786602828254
// MI455X (gfx1250) — compile-verified
//
#include <hip/hip_runtime.h>

typedef __attribute__((ext_vector_type(2))) float v2f;
typedef __attribute__((ext_vector_type(8))) float v8f;

#define BATCH   4
#define NPTS    8192
#define CIN     64
#define COUT    128
#define NPOINT  2048          // NPTS / 4
#define KNN     16
#define KRAW    67            // 3 + CIN
#define KPAD    68            // padded to multiple of 4 (17 wmma k-steps)
#define BNEPS   1e-5f

// ----------------------------------------------------------------------------
// 1) Furthest point sampling: one block per batch, dist in registers,
//    LDS tree argmax with first-index tie-break (matches jnp.argmax).
// ----------------------------------------------------------------------------
__global__ __launch_bounds__(1024) void fps_kernel(const float* __restrict__ xyz,
                                                   int* __restrict__ fps_idx)
{
    const int b   = blockIdx.x;
    const int tid = threadIdx.x;
    const float* xb = xyz + (size_t)b * NPTS * 3;

    __shared__ float sdist[1024];
    __shared__ int   sidx[1024];

    float dist[8];
#pragma unroll
    for (int i = 0; i < 8; ++i) dist[i] = 3.4e38f;

    int cur = 0;
    for (int s = 0; s < NPOINT; ++s) {
        if (tid == 0) fps_idx[b * NPOINT + s] = cur;
        const float cx = xb[cur * 3 + 0];
        const float cy = xb[cur * 3 + 1];
        const float cz = xb[cur * 3 + 2];

        float best = -1.0f; int bidx = 0;
#pragma unroll
        for (int i = 0; i < 8; ++i) {
            const int n = i * 1024 + tid;          // ascending per thread
            const float dx = xb[n * 3 + 0] - cx;
            const float dy = xb[n * 3 + 1] - cy;
            const float dz = xb[n * 3 + 2] - cz;
            const float d  = dx * dx + dy * dy + dz * dz;
            const float nd = fminf(dist[i], d);
            dist[i] = nd;
            if (nd > best) { best = nd; bidx = n; }  // strict > keeps first
        }
        sdist[tid] = best; sidx[tid] = bidx;
        __syncthreads();
        for (int off = 512; off > 0; off >>= 1) {
            if (tid < off) {
                const float d1 = sdist[tid],       d2 = sdist[tid + off];
                const int   i1 = sidx[tid],        i2 = sidx[tid + off];
                if (d2 > d1 || (d2 == d1 && i2 < i1)) { sdist[tid] = d2; sidx[tid] = i2; }
            }
            __syncthreads();
        }
        cur = sidx[0];
        __syncthreads();
    }
}

// ----------------------------------------------------------------------------
// 2) Gather new_xyz = xyz[fps_idx]
// ----------------------------------------------------------------------------
__global__ void gather_new_xyz_kernel(const float* __restrict__ xyz,
                                      const int* __restrict__ fps_idx,
                                      float* __restrict__ new_xyz)
{
    const int e = blockIdx.x * blockDim.x + threadIdx.x;
    if (e >= BATCH * NPOINT * 3) return;
    const int c = e % 3;
    const int p = (e / 3) % NPOINT;
    const int b = e / (3 * NPOINT);
    const int n = fps_idx[b * NPOINT + p];
    new_xyz[e] = xyz[((size_t)b * NPTS + n) * 3 + c];
}

// ----------------------------------------------------------------------------
// 3) kNN: one thread per query; register-resident sorted top-16
//    (branchless bubble insert, fully unrolled -> no scratch).
// ----------------------------------------------------------------------------
__global__ __launch_bounds__(128) void knn_kernel(const float* __restrict__ xyz,
                                                  const float* __restrict__ new_xyz,
                                                  int* __restrict__ knn_idx)
{
    const int q = blockIdx.x * blockDim.x + threadIdx.x;
    if (q >= BATCH * NPOINT) return;
    const int b = q / NPOINT;
    const float qx = new_xyz[q * 3 + 0];
    const float qy = new_xyz[q * 3 + 1];
    const float qz = new_xyz[q * 3 + 2];
    const float* xb = xyz + (size_t)b * NPTS * 3;

    float v[KNN]; int id[KNN];
#pragma unroll
    for (int t = 0; t < KNN; ++t) { v[t] = 3.4e38f; id[t] = 0; }

    for (int j = 0; j < NPTS; ++j) {
        if ((j & 63) == 0)
            __builtin_prefetch(xb + (size_t)(j + 256) * 3, 0, 1); // global_prefetch_b8
        const float dx = xb[j * 3 + 0] - qx;
        const float dy = xb[j * 3 + 1] - qy;
        const float dz = xb[j * 3 + 2] - qz;
        const float d  = dx * dx + dy * dy + dz * dz;
        if (d < v[KNN - 1]) {                     // strict: keep lower index on tie
            v[KNN - 1] = d; id[KNN - 1] = j;
#pragma unroll
            for (int t = KNN - 1; t > 0; --t) {
                if (v[t] < v[t - 1]) {
                    const float tv = v[t]; v[t] = v[t - 1]; v[t - 1] = tv;
                    const int   ti = id[t]; id[t] = id[t - 1]; id[t - 1] = ti;
                }
            }
        }
    }
#pragma unroll
    for (int t = 0; t < KNN; ++t) knn_idx[(size_t)q * KNN + t] = id[t];
}

// ----------------------------------------------------------------------------
// 4) Zero-pad W from 67 -> 68 rows
// ----------------------------------------------------------------------------
__global__ void pad_w_kernel(const float* __restrict__ W, float* __restrict__ Wpad)
{
    const int e = blockIdx.x * blockDim.x + threadIdx.x;
    if (e >= KPAD * COUT) return;
    const int r = e / COUT;
    Wpad[e] = (r < KRAW) ? W[e] : 0.0f;
}

// ----------------------------------------------------------------------------
// 5) Grouped-GEMM via V_WMMA_F32_16X16X4_F32.
//    One block per query point: M-tile = its 16 neighbors, 8 waves x 16 cols.
//    final_pass==0: emit per-block partial sum/sumsq per channel.
//    final_pass==1: BN(scale/shift) + ReLU + max over the 16 rows -> output.
// ----------------------------------------------------------------------------
__global__ __launch_bounds__(256) void group_gemm_kernel(
    const float* __restrict__ xyz, const float* __restrict__ features,
    const float* __restrict__ new_xyz, const int* __restrict__ knn_idx,
    const float* __restrict__ Wpad,
    const float* __restrict__ gamma, const float* __restrict__ beta,
    const float* __restrict__ meanv, const float* __restrict__ varv,
    float* __restrict__ partial_sum, float* __restrict__ partial_sq,
    float* __restrict__ out_features, int final_pass)
{
    const int q    = blockIdx.x;              // flattened (b, p)
    const int b    = q / NPOINT;
    const int tid  = threadIdx.x;
    const int lane = tid & 31;
    const int wv   = tid >> 5;                // wave id 0..7 -> N-tile

    __shared__ float As[KNN][KPAD];           // 16 x 68 grouped tile
    __shared__ float red[8][32];

    const float qx = new_xyz[q * 3 + 0];
    const float qy = new_xyz[q * 3 + 1];
    const float qz = new_xyz[q * 3 + 2];

    for (int e = tid; e < KNN * KPAD; e += 256) {
        const int kk = e / KPAD;
        const int c  = e % KPAD;
        const int n  = knn_idx[(size_t)q * KNN + kk];
        float val;
        if (c < 3) {
            const float pv = xyz[((size_t)b * NPTS + n) * 3 + c];
            val = pv - (c == 0 ? qx : (c == 1 ? qy : qz));
        } else if (c < KRAW) {
            val = features[((size_t)b * NPTS + n) * CIN + (c - 3)];
        } else {
            val = 0.0f;
        }
        As[kk][c] = val;
    }
    __syncthreads();

    const int col = wv * 16 + (lane & 15);    // output channel
    const int hi  = lane >> 4;                // lane half selects K pair
    const int row = lane & 15;                // A-matrix M = neighbor index

    v8f acc = {};
#pragma unroll
    for (int kb = 0; kb < KPAD / 4; ++kb) {
        const int k0 = kb * 4 + hi * 2;       // ISA 32-bit A/B lane layout
        v2f a, bb;
        a.x  = As[row][k0 + 0];
        a.y  = As[row][k0 + 1];
        bb.x = Wpad[(k0 + 0) * COUT + col];
        bb.y = Wpad[(k0 + 1) * COUT + col];
        acc = __builtin_amdgcn_wmma_f32_16x16x4_f32(
                  false, a, false, bb, (short)0, acc, false, false);
    }

    if (!final_pass) {
        float s = 0.0f, ss = 0.0f;
#pragma unroll
        for (int t = 0; t < 8; ++t) { const float h = acc[t]; s += h; ss += h * h; }
        red[wv][lane] = s;
        __syncthreads();
        float s2 = 0.0f;
        if (lane < 16) s2 = red[wv][lane] + red[wv][lane + 16];
        __syncthreads();
        red[wv][lane] = ss;
        __syncthreads();
        if (lane < 16) {
            const float ss2 = red[wv][lane] + red[wv][lane + 16];
            partial_sum[(size_t)q * COUT + col] = s2;
            partial_sq [(size_t)q * COUT + col] = ss2;
        }
    } else {
        const float mu    = meanv[col];
        const float scale = gamma[col] * rsqrtf(varv[col] + BNEPS);
        const float shift = beta[col] - mu * scale;
        float m = -3.4e38f;
#pragma unroll
        for (int t = 0; t < 8; ++t) {
            float h = acc[t] * scale + shift;
            h = fmaxf(h, 0.0f);                // ReLU
            m = fmaxf(m, h);                   // max over 8 of 16 neighbor rows
        }
        red[wv][lane] = m;
        __syncthreads();
        if (lane < 16) {
            const float mm = fmaxf(red[wv][lane], red[wv][lane + 16]);
            out_features[(size_t)q * COUT + col] = mm;
        }
    }
}

// ----------------------------------------------------------------------------
// 6) Deterministic per-channel reduction of partials -> mean / var
// ----------------------------------------------------------------------------
__global__ __launch_bounds__(256) void reduce_stats_kernel(
    const float* __restrict__ partial_sum, const float* __restrict__ partial_sq,
    float* __restrict__ meanv, float* __restrict__ varv)
{
    const int ch  = blockIdx.x;               // 0..127
    const int tid = threadIdx.x;
    __shared__ float ssum[256], ssq[256];
    float s = 0.0f, sq = 0.0f;
    for (int i = tid; i < BATCH * NPOINT; i += 256) {
        s  += partial_sum[(size_t)i * COUT + ch];
        sq += partial_sq [(size_t)i * COUT + ch];
    }
    ssum[tid] = s; ssq[tid] = sq;
    __syncthreads();
    for (int off = 128; off > 0; off >>= 1) {
        if (tid < off) { ssum[tid] += ssum[tid + off]; ssq[tid] += ssq[tid + off]; }
        __syncthreads();
    }
    if (tid == 0) {
        const float cnt = (float)(BATCH * NPOINT * KNN);
        const float mu  = ssum[0] / cnt;
        meanv[ch] = mu;
        varv[ch]  = ssq[0] / cnt - mu * mu;   // population var, matches jnp.var
    }
}

// ----------------------------------------------------------------------------
extern "C" void kernel_launch(void* const* d_in, const int* in_sizes, int n_in,
                              void* d_out, int out_size, void* d_ws, size_t ws_size,
                              hipStream_t stream)
{
    const float* xyz      = (const float*)d_in[0];
    const float* features = (const float*)d_in[1];
    const float* W        = (const float*)d_in[2];
    const float* gamma    = (const float*)d_in[3];
    const float* beta     = (const float*)d_in[4];

    float* out          = (float*)d_out;
    float* new_xyz      = out;                              // [B, NPOINT, 3]
    float* out_features = out + BATCH * NPOINT * 3;         // [B, NPOINT, COUT]

    char* ws = (char*)d_ws;
    size_t off = 0;
    int*   fps_idx     = (int*)(ws + off);   off += sizeof(int)   * BATCH * NPOINT;
    int*   knn_idx     = (int*)(ws + off);   off += sizeof(int)   * BATCH * NPOINT * KNN;
    float* Wpad        = (float*)(ws + off); off += sizeof(float) * KPAD * COUT;
    off = (off + 255) & ~(size_t)255;
    float* partial_sum = (float*)(ws + off); off += sizeof(float) * BATCH * NPOINT * COUT;
    float* partial_sq  = (float*)(ws + off); off += sizeof(float) * BATCH * NPOINT * COUT;
    float* meanv       = (float*)(ws + off); off += sizeof(float) * COUT;
    float* varv        = (float*)(ws + off); off += sizeof(float) * COUT;
    (void)ws_size; (void)in_sizes; (void)n_in; (void)out_size;

    fps_kernel<<<BATCH, 1024, 0, stream>>>(xyz, fps_idx);
    gather_new_xyz_kernel<<<(BATCH * NPOINT * 3 + 255) / 256, 256, 0, stream>>>(
        xyz, fps_idx, new_xyz);
    knn_kernel<<<(BATCH * NPOINT + 127) / 128, 128, 0, stream>>>(
        xyz, new_xyz, knn_idx);
    pad_w_kernel<<<(KPAD * COUT + 255) / 256, 256, 0, stream>>>(W, Wpad);
    group_gemm_kernel<<<BATCH * NPOINT, 256, 0, stream>>>(
        xyz, features, new_xyz, knn_idx, Wpad, gamma, beta,
        meanv, varv, partial_sum, partial_sq, out_features, 0);
    reduce_stats_kernel<<<COUT, 256, 0, stream>>>(
        partial_sum, partial_sq, meanv, varv);
    group_gemm_kernel<<<BATCH * NPOINT, 256, 0, stream>>>(
        xyz, features, new_xyz, knn_idx, Wpad, gamma, beta,
        meanv, varv, partial_sum, partial_sq, out_features, 1);
}